// Mixed_SelfAttention_43997644981170
// MI455X (gfx1250) — compile-verified
//
#include <hip/hip_runtime.h>
#include <hip/hip_bf16.h>

// MI455X / gfx1250 wave32 WMMA implementation, fp32 matrix path
// (V_WMMA_F32_16X16X4_F32) for full fp32 accuracy; problem is HBM-bandwidth
// dominated (~650MB traffic vs ~24 GFLOP), so fp32 WMMA is the right point.

typedef __attribute__((ext_vector_type(2))) float v2f;
typedef __attribute__((ext_vector_type(8))) float v8f;

#define NB 4
#define ND 128
#define NS 512
#define NT 32

__device__ __forceinline__ v8f wmma_f32_k4(v2f a, v2f b, v8f c) {
  // 8 args: (neg_a, A, neg_b, B, c_mod, C, reuse_a, reuse_b)
  return __builtin_amdgcn_wmma_f32_16x16x4_f32(
      false, a, false, b, (short)0, c, false, false);
}

// ---------------------------------------------------------------------------
// Kernel 1: q,k,v = xt @ W^T + b   (xt[b,t,s,d] = x[b,d,s,t])
// grid = (S/64, B*T), block = 256 (8 waves)
// wave w: s-tile (w&3)*16 within the 64-row chunk, e-range (w>>2)*64..+64
// ---------------------------------------------------------------------------
__global__ __launch_bounds__(256) void qkv_kernel(
    const float* __restrict__ x,
    const float* __restrict__ Wq, const float* __restrict__ bq,
    const float* __restrict__ Wk, const float* __restrict__ bk,
    const float* __restrict__ Wv, const float* __restrict__ bv,
    float* __restrict__ qkv_ws) {
  __shared__ float xt[64 * 132];  // [s_local][d], padded stride vs 64 banks

  const int bt = blockIdx.y;
  const int b = bt >> 5;   // bt / NT
  const int t = bt & 31;   // bt % NT
  const int s0 = blockIdx.x * 64;

  for (int i = threadIdx.x; i < 64 * 128; i += 256) {
    int sl = i >> 7, d = i & 127;
    xt[sl * 132 + d] =
        x[(((size_t)b * ND + d) * NS + (s0 + sl)) * NT + t];
  }
  __syncthreads();

  const int wave = threadIdx.x >> 5;
  const int lane = threadIdx.x & 31;
  const int half = lane >> 4;      // selects K pair {0,1} vs {2,3}
  const int lq = lane & 15;        // M (A) or N (B/C) index
  const int sl0 = (wave & 3) * 16;
  const int ebase = (wave >> 2) * 64;

  const float* Ws[3] = {Wq, Wk, Wv};
  const float* bs[3] = {bq, bk, bv};

  for (int m = 0; m < 3; ++m) {
    const float* __restrict__ W = Ws[m];
    const float* __restrict__ bias = bs[m];
    float* __restrict__ ob = qkv_ws + ((size_t)m * NB * NT + bt) * NS * ND;
    for (int e0 = ebase; e0 < ebase + 64; e0 += 16) {
      v8f c = {0.f, 0.f, 0.f, 0.f, 0.f, 0.f, 0.f, 0.f};
      for (int d0 = 0; d0 < 128; d0 += 4) {
        v2f a = *(const v2f*)&xt[(sl0 + lq) * 132 + d0 + 2 * half];
        v2f bf = *(const v2f*)&W[(size_t)(e0 + lq) * ND + d0 + 2 * half];
        c = wmma_f32_k4(a, bf, c);
      }
      float bb = bias[e0 + lq];
#pragma unroll
      for (int i = 0; i < 8; ++i) {
        int s = s0 + sl0 + 8 * half + i;
        ob[(size_t)s * ND + (e0 + lq)] = c[i] + bb;
      }
    }
  }
}

// ---------------------------------------------------------------------------
// Kernel 2: logits = q k^T ; scaled = logits*copula/sqrt(D+1e-7);
// weights = softmax(scaled). Emits original_logits + matrix_weights.
// grid = (S/16, B*T), block = 256 (8 waves). Each wave: 4 col-tiles of 16.
// ---------------------------------------------------------------------------
__global__ __launch_bounds__(256) void attn_logits_kernel(
    const float* __restrict__ qkv_ws, const float* __restrict__ copula,
    float* __restrict__ out_weights, float* __restrict__ out_logits) {
  __shared__ float qs[16 * 132];   // q tile
  __shared__ float sc[16 * 516];   // scaled logits (padded row stride)
  __shared__ float red[256];       // softmax reduction scratch

  const int bt = blockIdx.y;
  const int s0 = blockIdx.x * 16;
  const float* __restrict__ qb = qkv_ws + (size_t)bt * NS * ND;
  const float* __restrict__ kb = qkv_ws + ((size_t)(NB * NT + bt)) * NS * ND;

  for (int i = threadIdx.x; i < 16 * 128; i += 256) {
    int sl = i >> 7, d = i & 127;
    qs[sl * 132 + d] = qb[(size_t)(s0 + sl) * ND + d];
  }
  __syncthreads();

  const int wave = threadIdx.x >> 5;
  const int lane = threadIdx.x & 31;
  const int half = lane >> 4;
  const int lq = lane & 15;
  const float scale = rsqrtf(128.0f + 1e-7f);

  float* __restrict__ lgt = out_logits + (((size_t)bt) * NS + s0) * NS;
  const float* __restrict__ cop = copula + (((size_t)bt) * NS + s0) * NS;

  for (int nt = wave; nt < 32; nt += 8) {
    const int u0 = nt * 16;
    v8f c = {0.f, 0.f, 0.f, 0.f, 0.f, 0.f, 0.f, 0.f};
    for (int d0 = 0; d0 < 128; d0 += 4) {
      v2f a = *(const v2f*)&qs[lq * 132 + d0 + 2 * half];
      v2f bf = *(const v2f*)&kb[(size_t)(u0 + lq) * ND + d0 + 2 * half];
      c = wmma_f32_k4(a, bf, c);
    }
#pragma unroll
    for (int i = 0; i < 8; ++i) {
      int sl = 8 * half + i;
      int u = u0 + lq;
      float l = c[i];
      lgt[(size_t)sl * NS + u] = l;
      sc[sl * 516 + u] = l * cop[(size_t)sl * NS + u] * scale;
    }
  }
  __syncthreads();

  // softmax: 16 threads per row, 16 rows
  const int row = threadIdx.x >> 4;
  const int col = threadIdx.x & 15;
  float m = -INFINITY;
  for (int u = col; u < NS; u += 16) m = fmaxf(m, sc[row * 516 + u]);
  red[threadIdx.x] = m;
  __syncthreads();
  if (col == 0) {
    float mm = red[row * 16];
    for (int j = 1; j < 16; ++j) mm = fmaxf(mm, red[row * 16 + j]);
    red[row * 16] = mm;
  }
  __syncthreads();
  const float rowmax = red[row * 16];
  __syncthreads();  // everyone has read rowmax before red is reused
  float ssum = 0.f;
  for (int u = col; u < NS; u += 16) {
    float e = __expf(sc[row * 516 + u] - rowmax);
    sc[row * 516 + u] = e;
    ssum += e;
  }
  red[threadIdx.x] = ssum;
  __syncthreads();
  if (col == 0) {
    float s = 0.f;
    for (int j = 0; j < 16; ++j) s += red[row * 16 + j];
    red[row * 16] = s;
  }
  __syncthreads();
  const float inv = 1.0f / red[row * 16];
  float* __restrict__ wb = out_weights + (((size_t)bt) * NS + s0) * NS;
  for (int u = col; u < NS; u += 16)
    wb[(size_t)row * NS + u] = sc[row * 516 + u] * inv;
}

// ---------------------------------------------------------------------------
// Kernel 3: attention_output = weights @ v
// grid = (S/16, B*T), block = 256 (8 waves); wave w: e-tile w*16, K over u.
// ---------------------------------------------------------------------------
__global__ __launch_bounds__(256) void attn_av_kernel(
    const float* __restrict__ qkv_ws, const float* __restrict__ weights,
    float* __restrict__ out) {
  __shared__ float wls[16 * 516];  // weights tile, padded stride

  const int bt = blockIdx.y;
  const int s0 = blockIdx.x * 16;
  const float* __restrict__ vb =
      qkv_ws + ((size_t)(2 * NB * NT + bt)) * NS * ND;
  const float* __restrict__ wb = weights + (((size_t)bt) * NS + s0) * NS;

  for (int i = threadIdx.x; i < 16 * 512; i += 256) {
    int sl = i >> 9, u = i & 511;
    wls[sl * 516 + u] = wb[(size_t)sl * NS + u];
  }
  __syncthreads();

  const int wave = threadIdx.x >> 5;
  const int lane = threadIdx.x & 31;
  const int half = lane >> 4;
  const int lq = lane & 15;
  const int e0 = wave * 16;

  v8f c = {0.f, 0.f, 0.f, 0.f, 0.f, 0.f, 0.f, 0.f};
  for (int u0 = 0; u0 < NS; u0 += 4) {
    v2f a = *(const v2f*)&wls[lq * 516 + u0 + 2 * half];
    v2f bf;
    bf.x = vb[(size_t)(u0 + 2 * half) * ND + e0 + lq];
    bf.y = vb[(size_t)(u0 + 2 * half + 1) * ND + e0 + lq];
    c = wmma_f32_k4(a, bf, c);
  }
  float* __restrict__ ob = out + (((size_t)bt) * NS + s0) * ND;
#pragma unroll
  for (int i = 0; i < 8; ++i) {
    int sl = 8 * half + i;
    ob[(size_t)sl * ND + e0 + lq] = c[i];
  }
}

// ---------------------------------------------------------------------------
extern "C" void kernel_launch(void* const* d_in, const int* in_sizes, int n_in,
                              void* d_out, int out_size, void* d_ws,
                              size_t ws_size, hipStream_t stream) {
  (void)in_sizes; (void)n_in; (void)out_size; (void)ws_size;
  const float* x   = (const float*)d_in[0];
  const float* cop = (const float*)d_in[1];
  const float* Wq  = (const float*)d_in[2];
  const float* bq  = (const float*)d_in[3];
  const float* Wk  = (const float*)d_in[4];
  const float* bk  = (const float*)d_in[5];
  const float* Wv  = (const float*)d_in[6];
  const float* bv  = (const float*)d_in[7];

  float* out    = (float*)d_out;
  float* attn   = out;                                   // [B,T,S,D]
  float* wts    = attn + (size_t)NB * NT * NS * ND;      // [B,T,S,S]
  float* logits = wts + (size_t)NB * NT * NS * NS;       // [B,T,S,S]
  float* qkv    = (float*)d_ws;  // q|k|v each [B,T,S,D] = 100.7MB total

  dim3 blk(256);
  dim3 g1(NS / 64, NB * NT);
  qkv_kernel<<<g1, blk, 0, stream>>>(x, Wq, bq, Wk, bk, Wv, bv, qkv);

  dim3 g2(NS / 16, NB * NT);
  attn_logits_kernel<<<g2, blk, 0, stream>>>(qkv, cop, wts, logits);
  attn_av_kernel<<<g2, blk, 0, stream>>>(qkv, wts, attn);
}